// DeepECCNet_1176821039625
// MI455X (gfx1250) — compile-verified
//
#include <hip/hip_runtime.h>
#include <math.h>

// ---------------- problem constants (from reference) ----------------
#define N_NODES 100000
#define N_EDGES 1600000
#define HID     128
#define IN_NODE 129
#define H2      64

typedef __attribute__((ext_vector_type(2))) float v2f;
typedef __attribute__((ext_vector_type(8))) float v8f;

__device__ __forceinline__ float sigmoidf(float v) {
    return 1.0f / (1.0f + __expf(-v));
}

// ---------------- utility: fill ----------------
__global__ void fill_kernel(float* __restrict__ p, float v, int n) {
    int i = blockIdx.x * blockDim.x + threadIdx.x;
    if (i < n) p[i] = v;
}

// ---------------- degree: deg[i] = 1 (self loop) + #edges with tgt==i ----------------
__global__ void deg_kernel(const long long* __restrict__ tgt, float* __restrict__ deg) {
    int e = blockIdx.x * blockDim.x + threadIdx.x;
    if (e < N_EDGES) atomicAdd(&deg[(int)tgt[e]], 1.0f);
}

// ---------------- t-head: t = sigmoid(relu(x@t1W+t1b)@t2W + t2b) ----------------
// one 64-thread block per node; thread j owns hidden unit j
__global__ void tpath_kernel(const float* __restrict__ x,
                             const float* __restrict__ t1W, const float* __restrict__ t1b,
                             const float* __restrict__ t2W, const float* __restrict__ t2b,
                             float* __restrict__ t) {
    __shared__ float red[H2];
    int node = blockIdx.x;
    int j = threadIdx.x;                       // 0..63
    const float* xr = x + (long)node * IN_NODE;
    float s = t1b[j];
    for (int k = 0; k < IN_NODE; ++k)
        s = fmaf(xr[k], t1W[k * H2 + j], s);   // broadcast xr[k], coalesced t1W row
    s = fmaxf(s, 0.0f);
    red[j] = s * t2W[j];
    __syncthreads();
    if (j == 0) {
        float acc = t2b[0];
        #pragma unroll
        for (int q = 0; q < H2; ++q) acc += red[q];
        t[node] = sigmoidf(acc);
    }
}

// ---------------- WMMA fp32 GEMM: out[16xHID strip] = A[16xK=128] @ W[128xHID] + bias ----
// block = 256 threads = 8 wave32; wave w computes 16x16 tile at column 16*w.
// A strip staged in LDS (row pitch 132 floats -> bank-conflict free for the
// stride-128 column reads of the A fragment).
__global__ void gemm_wmma_kernel(const float* __restrict__ A, int lda, int colOff,
                                 const float* __restrict__ W,
                                 const float* __restrict__ bias,
                                 float* __restrict__ out) {
    __shared__ float lA[16 * 132];
    const int node0 = blockIdx.x * 16;

    // cooperative load of the 16x128 A strip (handles layer-0 x[:,1:] via lda/colOff)
    for (int idx = threadIdx.x; idx < 16 * 128; idx += blockDim.x) {
        int r = idx >> 7, c = idx & 127;
        lA[r * 132 + c] = A[(long)(node0 + r) * lda + colOff + c];
    }
    __syncthreads();

    const int lane  = threadIdx.x & 31;
    const int wave  = threadIdx.x >> 5;        // 0..7
    const int colT  = wave * 16;
    const int m     = lane & 15;               // M (for A) / N (for B)
    const int khalf = (lane >> 4) << 1;        // lanes 0-15 -> K+{0,1}; 16-31 -> K+{2,3}

    v8f c = {};
    for (int k = 0; k < 128; k += 4) {
        v2f a, b;
        // A 16x4 f32 fragment (ISA 7.12.2): VGPR0 = K even half, VGPR1 = K odd
        a.x = lA[m * 132 + k + khalf];
        a.y = lA[m * 132 + k + khalf + 1];
        // B 4x16 f32 fragment: lanes 0-15 rows K,K+1 ; lanes 16-31 rows K+2,K+3
        b.x = W[(k + khalf)     * HID + colT + m];
        b.y = W[(k + khalf + 1) * HID + colT + m];
        // (neg_a, A, neg_b, B, c_mod, C, reuse_a, reuse_b)
        c = __builtin_amdgcn_wmma_f32_16x16x4_f32(false, a, false, b,
                                                  (short)0, c, false, false);
    }

    // C/D layout: VGPR r holds M=r (lanes 0-15) and M=r+8 (lanes 16-31), N = lane&15
    const int mbase = (lane >> 4) * 8;
    const float bcol = bias[colT + m];
    #pragma unroll
    for (int r = 0; r < 8; ++r) {
        out[(long)(node0 + mbase + r) * HID + colT + m] = c[r] + bcol;
    }
}

// ---------------- edge scatter: agg[tgt] += sigmoid(ea*w) * ht[src] ----------------
// one wave32 per edge; each lane moves a float4 (128 floats / edge)
__global__ void edge_kernel(const long long* __restrict__ src,
                            const long long* __restrict__ tgt,
                            const float* __restrict__ ea,
                            const float* __restrict__ w,
                            const float* __restrict__ ht,
                            float* __restrict__ agg) {
    long long gid = (long long)blockIdx.x * blockDim.x + threadIdx.x;
    int e    = (int)(gid >> 5);
    int lane = (int)(gid & 31);
    if (e >= N_EDGES) return;
    float gate = sigmoidf(ea[e] * w[0]);
    int s = (int)src[e];
    int d = (int)tgt[e];
    float4 v = ((const float4*)(ht + (long)s * HID))[lane];   // L2-resident gather
    float* ap = agg + (long)d * HID + lane * 4;
    atomicAdd(ap + 0, gate * v.x);
    atomicAdd(ap + 1, gate * v.y);
    atomicAdd(ap + 2, gate * v.z);
    atomicAdd(ap + 3, gate * v.w);
}

// ---------------- finalize: h = relu((agg + sigmoid(w)*ht) / deg) ----------------
// self-loop contribution (ea==1 -> gate = sigmoid(w)) folded in here: no atomics needed
__global__ void finalize_kernel(const float* __restrict__ agg,
                                const float* __restrict__ ht,
                                const float* __restrict__ deg,
                                const float* __restrict__ w,
                                float* __restrict__ h) {
    long long i = (long long)blockIdx.x * blockDim.x + threadIdx.x;
    if (i >= (long long)N_NODES * HID) return;
    int node = (int)(i >> 7);
    float gs = sigmoidf(w[0]);
    float v = (agg[i] + gs * ht[i]) / deg[node];
    h[i] = fmaxf(v, 0.0f);
}

// ---------------- output head: out = sigmoid(relu([h,t]@l1W+l1b)@l2W + l2b) --------
__global__ void head_kernel(const float* __restrict__ h, const float* __restrict__ t,
                            const float* __restrict__ l1W, const float* __restrict__ l1b,
                            const float* __restrict__ l2W, const float* __restrict__ l2b,
                            float* __restrict__ out) {
    __shared__ float red[H2];
    int node = blockIdx.x;
    int j = threadIdx.x;                       // 0..63
    const float* hr = h + (long)node * HID;
    float s = l1b[j];
    for (int k = 0; k < HID; ++k)
        s = fmaf(hr[k], l1W[k * H2 + j], s);
    s = fmaf(t[node], l1W[HID * H2 + j], s);   // row 128 of lin1_W handles the t column
    s = fmaxf(s, 0.0f);
    red[j] = s * l2W[j];
    __syncthreads();
    if (j == 0) {
        float acc = l2b[0];
        #pragma unroll
        for (int q = 0; q < H2; ++q) acc += red[q];
        out[node] = sigmoidf(acc);
    }
}

// =====================================================================
extern "C" void kernel_launch(void* const* d_in, const int* in_sizes, int n_in,
                              void* d_out, int out_size, void* d_ws, size_t ws_size,
                              hipStream_t stream) {
    // setup_inputs() dict order, tuples flattened:
    const float*     x    = (const float*)d_in[0];
    const long long* eidx = (const long long*)d_in[1];   // int64 per reference
    const float*     ea   = (const float*)d_in[2];
    const float* convW[3] = {(const float*)d_in[3], (const float*)d_in[4], (const float*)d_in[5]};
    const float* convB[3] = {(const float*)d_in[6], (const float*)d_in[7], (const float*)d_in[8]};
    const float* edgeW[3] = {(const float*)d_in[9], (const float*)d_in[10], (const float*)d_in[11]};
    const float* l1W = (const float*)d_in[12];
    const float* l1b = (const float*)d_in[13];
    const float* l2W = (const float*)d_in[14];
    const float* l2b = (const float*)d_in[15];
    const float* t1W = (const float*)d_in[16];
    const float* t1b = (const float*)d_in[17];
    const float* t2W = (const float*)d_in[18];
    const float* t2b = (const float*)d_in[19];

    const long long* src = eidx;             // edge_index[0]
    const long long* tgt = eidx + N_EDGES;   // edge_index[1]

    // workspace carve-out (floats): deg | t | ht | agg | h  == (2 + 3*128)*N ~ 154 MB
    float* ws  = (float*)d_ws;
    float* deg = ws;
    float* t   = ws + N_NODES;
    float* ht  = ws + 2 * (size_t)N_NODES;
    float* agg = ht + (size_t)N_NODES * HID;
    float* h   = agg + (size_t)N_NODES * HID;

    const int NH = N_NODES * HID;            // 12.8M, fits int

    // 1) degrees (self loop = 1.0 baseline)
    fill_kernel<<<(N_NODES + 255) / 256, 256, 0, stream>>>(deg, 1.0f, N_NODES);
    deg_kernel<<<(N_EDGES + 255) / 256, 256, 0, stream>>>(tgt, deg);

    // 2) t head
    tpath_kernel<<<N_NODES, H2, 0, stream>>>(x, t1W, t1b, t2W, t2b, t);

    // 3) three ECC layers
    for (int L = 0; L < 3; ++L) {
        fill_kernel<<<(NH + 255) / 256, 256, 0, stream>>>(agg, 0.0f, NH);
        if (L == 0) {
            // h0 = x[:, 1:]  -> read x with row stride 129, column offset 1
            gemm_wmma_kernel<<<N_NODES / 16, 256, 0, stream>>>(x, IN_NODE, 1,
                                                               convW[0], convB[0], ht);
        } else {
            gemm_wmma_kernel<<<N_NODES / 16, 256, 0, stream>>>(h, HID, 0,
                                                               convW[L], convB[L], ht);
        }
        edge_kernel<<<(N_EDGES * 32) / 256, 256, 0, stream>>>(src, tgt, ea, edgeW[L], ht, agg);
        finalize_kernel<<<(NH + 255) / 256, 256, 0, stream>>>(agg, ht, deg, edgeW[L], h);
    }

    // 4) output head
    head_kernel<<<N_NODES, H2, 0, stream>>>(h, t, l1W, l1b, l2W, l2b, (float*)d_out);
}